// Pad_86363202388359
// MI455X (gfx1250) — compile-verified
//
#include <hip/hip_runtime.h>
#include <hip/hip_bf16.h>

// Ragged -> padded copy: (sum(seqlen)=8192, 4096) f32 -> (8, 2048, 4096) f32,
// zero-filled tails. Pure HBM-bandwidth problem (~384 MiB total traffic,
// ~16.5 us floor at 23.3 TB/s). We use the CDNA5 async Global<->LDS DMA path
// (GLOBAL_LOAD_ASYNC_TO_LDS_B128 / GLOBAL_STORE_ASYNC_FROM_LDS_B128, ASYNCcnt)
// so each wave keeps many 128b-per-lane transfers in flight with zero VGPR
// data-register pressure.

#define PAD_BATCH   8
#define PAD_MAXSEQ  2048
#define PAD_HIDDEN  4096                 // floats per row
#define ROW_F4      (PAD_HIDDEN / 4)     // 1024 float4 per row
#define BLOCK_T     256                  // 8 waves (wave32)
#define CHUNKS      (ROW_F4 / BLOCK_T)   // 4 x b128 per thread

// Builtin signature (from hipcc diagnostics):
//   param0 = generic ptr to int __vector_size__(16)
//   param1 = addrspace(3) ptr to int __vector_size__(16)
typedef int v4i __attribute__((__vector_size__(16)));
typedef __attribute__((address_space(3))) v4i lds_v4i;

__global__ __launch_bounds__(BLOCK_T)
void pad_ragged_kernel(const float* __restrict__ in,
                       const int*   __restrict__ seqlen,
                       float*       __restrict__ out) {
    __shared__ float4 stage[ROW_F4];     // 16 KiB staging buffer

    const int row = blockIdx.x;          // 0 .. BATCH*MAXSEQ-1
    const int b   = row >> 11;           // row / 2048
    const int pos = row & (PAD_MAXSEQ - 1);

    // Per-batch start offset = prefix sum of seqlen[0..b-1]; uniform scalar work.
    int off = 0;
    int len = 0;
#pragma unroll
    for (int i = 0; i < PAD_BATCH; ++i) {
        const int l = seqlen[i];
        if (i < b) off += l;
        if (i == b) len = l;
    }

    float4* dst = (float4*)out + (size_t)row * ROW_F4;
    const int t = threadIdx.x;

    if (pos < len) {
        // ---- valid row: async DMA global -> LDS -> global ----
        const float4* src = (const float4*)in + (size_t)(off + pos) * ROW_F4;

#pragma unroll
        for (int k = 0; k < CHUNKS; ++k) {
            const int idx = t + k * BLOCK_T;    // lane-contiguous 16B chunks
            __builtin_amdgcn_global_load_async_to_lds_b128(
                (v4i*)(src + idx),
                (lds_v4i*)(void*)(stage + idx),
                /*offset=*/0, /*cpol=*/0);
        }

        // Wait for this wave's async loads: LDS bytes each lane will store
        // were written by that same lane's loads, so ASYNCcnt==0 suffices
        // (no cross-wave sharing -> no workgroup barrier needed).
        asm volatile("s_wait_asynccnt 0" ::: "memory");

#pragma unroll
        for (int k = 0; k < CHUNKS; ++k) {
            const int idx = t + k * BLOCK_T;
            __builtin_amdgcn_global_store_async_from_lds_b128(
                (v4i*)(dst + idx),
                (lds_v4i*)(void*)(stage + idx),
                /*offset=*/0, /*cpol=*/0);
        }

        // Drain stores (S_ENDPGM also implies wait-idle; explicit for clarity).
        asm volatile("s_wait_asynccnt 0" ::: "memory");
    } else {
        // ---- padding row: pure b128 zero stores, no read traffic ----
        const float4 z = make_float4(0.f, 0.f, 0.f, 0.f);
#pragma unroll
        for (int k = 0; k < CHUNKS; ++k) {
            dst[t + k * BLOCK_T] = z;
        }
    }
}

extern "C" void kernel_launch(void* const* d_in, const int* in_sizes, int n_in,
                              void* d_out, int out_size, void* d_ws, size_t ws_size,
                              hipStream_t stream) {
    (void)in_sizes; (void)n_in; (void)out_size; (void)d_ws; (void)ws_size;
    const float* in     = (const float*)d_in[0];   // (8192, 4096) f32
    const int*   seqlen = (const int*)d_in[1];     // (8,) i32
    float*       out    = (float*)d_out;           // (8, 2048, 4096) f32

    const dim3 grid(PAD_BATCH * PAD_MAXSEQ);       // one block per output row
    const dim3 block(BLOCK_T);
    pad_ragged_kernel<<<grid, block, 0, stream>>>(in, seqlen, out);
}